// GNCA_78211354460679
// MI455X (gfx1250) — compile-verified
//
#include <hip/hip_runtime.h>
#include <math.h>

#define NP 8192
#define R2   (0.05f * 0.05f)
#define CR2  (0.10f * 0.10f)       // (2*RADIUS)^2
#define ACC_SCALE 0.005f
#define MAX_VEL 0.05f
#define EPSV 1e-6f
#define TILE_J 2048
#define JSPLIT 4                   // j-dimension split across blocks (occupancy)

typedef __attribute__((ext_vector_type(2))) float v2f;
typedef __attribute__((ext_vector_type(8))) float v8f;

// ws layout (floats):
//   [0,      NP*4)  y = x @ w_neigh
//   [NP*4,   NP*8)  s = x @ w_self + b
//   [NP*8,   NP*16) per-row partial accumulators: a0..a3, vis, cd, cs, pad

// Kernel 1: small GEMMs + zero all accumulators (runs every launch; harness
// does not re-zero buffers between replays).
__global__ __launch_bounds__(256) void gnca_prep(
    const float* __restrict__ x, const float* __restrict__ w_self,
    const float* __restrict__ w_neigh, const float* __restrict__ b,
    float* __restrict__ ws, float* __restrict__ out)
{
  int i = blockIdx.x * 256 + threadIdx.x;
  if (i < 6) out[NP * 7 + i] = 0.0f;   // vel_bonus[2], border, food, dead, visible
  if (i >= NP) return;
  float xv[7];
#pragma unroll
  for (int k = 0; k < 7; ++k) xv[k] = x[(size_t)i * 7 + k];
#pragma unroll
  for (int c = 0; c < 4; ++c) {
    float y = 0.0f, s = b[c];
#pragma unroll
    for (int k = 0; k < 7; ++k) {
      y = fmaf(xv[k], w_neigh[k * 4 + c], y);
      s = fmaf(xv[k], w_self[k * 4 + c], s);
    }
    ws[(size_t)i * 4 + c] = y;
    ws[(size_t)NP * 4 + (size_t)i * 4 + c] = s;
  }
  float* ar = ws + (size_t)NP * 8 + (size_t)i * 8;
#pragma unroll
  for (int k = 0; k < 8; ++k) ar[k] = 0.0f;
}

// Kernel 2: all-pairs. WMMA computes dist^2 directly:
//   A row i = [-2px_i, -2py_i, 1, r2_i], B col j = [px_j, py_j, r2_j, 1]
//   => D[i][j] = r2_i + r2_j - 2 p_i.p_j
// Diagonal is intentionally counted; exact self term subtracted in finalize.
__global__ __launch_bounds__(256) void gnca_pair(
    const float* __restrict__ x, float* __restrict__ ws)
{
  __shared__ float sPos[TILE_J * 2];
  __shared__ float sY[TILE_J * 4];
  __shared__ float sFlag[TILE_J];

  const int lane = threadIdx.x & 31;
  const int wv   = threadIdx.x >> 5;
  const int half = lane >> 4;          // 0: lanes 0-15 (K0,K1), 1: lanes 16-31 (K2,K3)
  const int lcol = lane & 15;
  const int cb    = (blockIdx.x % JSPLIT) * TILE_J;          // my j slice
  const int ibase = ((blockIdx.x / JSPLIT) * 8 + wv) * 16;   // my 16 rows

  // Stage this block's j slice into LDS (pos, message vec y, cell flag).
  for (int t = threadIdx.x; t < TILE_J; t += 256) {
    const float* xr = x + (size_t)(cb + t) * 7;
    sPos[t * 2 + 0] = xr[0];
    sPos[t * 2 + 1] = xr[1];
    sFlag[t] = (xr[4] > 0.5f) ? 1.0f : 0.0f;
    const float* yr = ws + (size_t)(cb + t) * 4;
    sY[t * 4 + 0] = yr[0]; sY[t * 4 + 1] = yr[1];
    sY[t * 4 + 2] = yr[2]; sY[t * 4 + 3] = yr[3];
  }

  // A operand (16x4 f32): lanes 0-15 carry K0,K1 = -2*pos; lanes 16-31 carry
  // K2 = 1.0, K3 = r2_i of row lcol.
  v2f aop;
  {
    const float* xr = x + (size_t)(ibase + lcol) * 7;
    float px = xr[0], py = xr[1];
    float r2i = px * px + py * py;
    aop[0] = half ? 1.0f : (-2.0f * px);
    aop[1] = half ? r2i  : (-2.0f * py);
  }

  float a0[8], a1[8], a2[8], a3[8], vis[8], cd[8], cs[8];
#pragma unroll
  for (int v = 0; v < 8; ++v) { a0[v]=a1[v]=a2[v]=a3[v]=vis[v]=cd[v]=cs[v]=0.0f; }

  __syncthreads();

  for (int jt = 0; jt < TILE_J; jt += 16) {
    int jl = jt + lcol;                 // my column within the 16-wide j tile
    float pjx = sPos[jl * 2 + 0], pjy = sPos[jl * 2 + 1];
    float cj = sFlag[jl], fj = 1.0f - cj;
    float y0 = sY[jl * 4 + 0], y1 = sY[jl * 4 + 1];
    float y2 = sY[jl * 4 + 2], y3 = sY[jl * 4 + 3];
    float r2j = pjx * pjx + pjy * pjy;

    v2f bop;                            // B (4x16): K0,K1 = pos_j; K2 = r2_j; K3 = 1
    bop[0] = half ? r2j  : pjx;
    bop[1] = half ? 1.0f : pjy;

    v8f d2 = {};
    d2 = __builtin_amdgcn_wmma_f32_16x16x4_f32(
        false, aop, false, bop, (short)0, d2, false, false);

#pragma unroll
    for (int v = 0; v < 8; ++v) {
      float m   = (d2[v] < R2)  ? 1.0f : 0.0f;   // adjacency (diag included)
      float m2c = (d2[v] < CR2) ? cj   : 0.0f;   // consume-radius cell count
      a0[v] = fmaf(m, y0, a0[v]);  a1[v] = fmaf(m, y1, a1[v]);
      a2[v] = fmaf(m, y2, a2[v]);  a3[v] = fmaf(m, y3, a3[v]);
      vis[v] = fmaf(m, fj, vis[v]);      // visible-food pairs
      cd[v]  = fmaf(m, cj, cd[v]);       // cell-neighbor degree
      cs[v] += m2c;                      // consume degree (cells within 2R)
    }
  }

  // Width-16 butterfly: sum partials across the 16 lanes sharing each row.
#pragma unroll
  for (int v = 0; v < 8; ++v) {
#pragma unroll
    for (int off = 8; off >= 1; off >>= 1) {
      a0[v] += __shfl_xor(a0[v], off, 16);
      a1[v] += __shfl_xor(a1[v], off, 16);
      a2[v] += __shfl_xor(a2[v], off, 16);
      a3[v] += __shfl_xor(a3[v], off, 16);
      vis[v] += __shfl_xor(vis[v], off, 16);
      cd[v]  += __shfl_xor(cd[v],  off, 16);
      cs[v]  += __shfl_xor(cs[v],  off, 16);
    }
  }

  if (lcol == 0) {                      // lanes 0 and 16 each commit 8 rows
    float* wsAcc = ws + (size_t)NP * 8;
#pragma unroll
    for (int v = 0; v < 8; ++v) {
      int ir = ibase + v + 8 * half;    // C-tile VGPR v holds row v + 8*half
      float* ar = wsAcc + (size_t)ir * 8;
      atomicAdd(ar + 0, a0[v]);  atomicAdd(ar + 1, a1[v]);
      atomicAdd(ar + 2, a2[v]);  atomicAdd(ar + 3, a3[v]);
      atomicAdd(ar + 4, vis[v]); atomicAdd(ar + 5, cd[v]);
      atomicAdd(ar + 6, cs[v]);
    }
  }
}

// Kernel 3: O(N) finalize — subtract self term, state update, reductions.
__global__ __launch_bounds__(256) void gnca_final(
    const float* __restrict__ x, const float* __restrict__ ws,
    float* __restrict__ out)
{
  int i = blockIdx.x * 256 + threadIdx.x;   // grid is exactly NP threads
  const float* ar = ws + (size_t)NP * 8 + (size_t)i * 8;
  float A0 = ar[0], A1 = ar[1], A2 = ar[2], A3 = ar[3];
  float VIS = ar[4], CD = ar[5], CS = ar[6];

  const float* xr = x + (size_t)i * 7;
  float px = xr[0], py = xr[1], vx = xr[2], vy = xr[3], typ = xr[4];
  float ci = (typ > 0.5f) ? 1.0f : 0.0f;

  // Remove the diagonal (self) contribution counted by the pair kernel.
  const float* yi = ws + (size_t)i * 4;
  A0 -= yi[0]; A1 -= yi[1]; A2 -= yi[2]; A3 -= yi[3];
  VIS -= (1.0f - ci);
  CD  -= ci;
  CS  -= ci;

  const float* sr = ws + (size_t)NP * 4 + (size_t)i * 4;
  float sc = ACC_SCALE * ci;
  float h0 = (sr[0] + A0) * sc, h1 = (sr[1] + A1) * sc;
  float h2 = (sr[2] + A2) * sc, h3 = (sr[3] + A3) * sc;
  float nvx = fminf(fmaxf(vx + h0, -MAX_VEL), MAX_VEL);
  float nvy = fminf(fmaxf(vy + h1, -MAX_VEL), MAX_VEL);
  float npx = px + nvx, npy = py + nvy;
  float bc = 0.0f;
  if (fabsf(npx) > 1.0f) bc += logf(fabsf(npx) + EPSV);
  if (fabsf(npy) > 1.0f) bc += logf(fabsf(npy) + EPSV);
  bool dead     = (ci > 0.5f) && (CD < 2.5f);   // cell_deg < 3
  bool consumed = (ci < 0.5f) && (CS > 4.5f);   // consume_deg >= 5
  float keep = (dead || consumed) ? 0.0f : 1.0f;

  float* orow = out + (size_t)i * 7;
  orow[0] = npx * keep; orow[1] = npy * keep;
  orow[2] = nvx * keep; orow[3] = nvy * keep;
  orow[4] = typ * keep; orow[5] = h2 * keep; orow[6] = h3 * keep;

  // Wave-level reduction of the 6 scalar outputs, then one atomic per wave.
  float sv0 = fabsf(nvx) * (1.0f / NP);
  float sv1 = fabsf(nvy) * (1.0f / NP);
  float sv2 = bc;
  float sv3 = consumed ? 1.0f : 0.0f;
  float sv4 = dead ? 1.0f : 0.0f;
  float sv5 = VIS;
#pragma unroll
  for (int off = 16; off >= 1; off >>= 1) {
    sv0 += __shfl_xor(sv0, off, 32);
    sv1 += __shfl_xor(sv1, off, 32);
    sv2 += __shfl_xor(sv2, off, 32);
    sv3 += __shfl_xor(sv3, off, 32);
    sv4 += __shfl_xor(sv4, off, 32);
    sv5 += __shfl_xor(sv5, off, 32);
  }
  if ((threadIdx.x & 31) == 0) {
    atomicAdd(out + NP * 7 + 0, sv0);
    atomicAdd(out + NP * 7 + 1, sv1);
    atomicAdd(out + NP * 7 + 2, sv2);
    atomicAdd(out + NP * 7 + 3, sv3);
    atomicAdd(out + NP * 7 + 4, sv4);
    atomicAdd(out + NP * 7 + 5, sv5);
  }
}

extern "C" void kernel_launch(void* const* d_in, const int* in_sizes, int n_in,
                              void* d_out, int out_size, void* d_ws, size_t ws_size,
                              hipStream_t stream) {
  const float* x       = (const float*)d_in[0];
  const float* w_self  = (const float*)d_in[1];
  const float* w_neigh = (const float*)d_in[2];
  const float* b       = (const float*)d_in[3];
  float* out = (float*)d_out;
  float* ws  = (float*)d_ws;   // needs NP*16 floats = 512 KB

  gnca_prep <<<NP / 256, 256, 0, stream>>>(x, w_self, w_neigh, b, ws, out);
  gnca_pair <<<(NP / 128) * JSPLIT, 256, 0, stream>>>(x, ws);   // 256 blocks, 2048 waves
  gnca_final<<<NP / 256, 256, 0, stream>>>(x, ws, out);
}